// AR_73607149519056
// MI455X (gfx1250) — compile-verified
//
#include <hip/hip_runtime.h>
#include <math.h>

typedef float v2f __attribute__((ext_vector_type(2)));
typedef float v8f __attribute__((ext_vector_type(8)));

#define AR_P      128                       // filter length (fixed by reference)
#define TILE_OUT  256                       // outputs per wave (one 16x16 WMMA tile)
#define WAVES_PB  8                         // wave32 waves per 256-thread block
#define BLOCK_OUT (TILE_OUT * WAVES_PB)     // 2048 outputs per block
#define Y_STAGE   (BLOCK_OUT + AR_P)        // 2176 floats of y staged in LDS
#define KCHUNKS   ((AR_P + 16) / 4)         // 36 WMMA K-steps (K_total = 144)
#define PP_LEN    (AR_P + 32)               // zero-padded params band source

// ---------------------------------------------------------------------------
// Kernel 1: pred = Toeplitz(y) @ params via V_WMMA_F32_16X16X4_F32.
//   D[m,nn] = pred[base + m + 16*nn] = sum_k A[m,k] * B[k,nn]
//   A[m,k]  = params[k-m]  (banded, 0 <= k-m < 128), invariant across tiles
//   B[k,nn] = y[base + 16*nn + k]
// ---------------------------------------------------------------------------
__global__ __launch_bounds__(256) void AR_wmma_pred(const float* __restrict__ y,
                                                    const float* __restrict__ params,
                                                    float* __restrict__ pred,
                                                    int n, int N) {
    __shared__ float ylds[Y_STAGE];
    __shared__ float pp[PP_LEN];            // pp[i] = params[i-16] (zero-padded)

    const int tid       = threadIdx.x;
    const int blockBase = blockIdx.x * BLOCK_OUT;

    // Stage y window (clamped; clamped values only feed outputs >= n).
    for (int i = tid; i < Y_STAGE; i += 256) {
        int g = blockBase + i;
        if (g >= N) g = N - 1;
        ylds[i] = y[g];
    }
    // Stage zero-padded params for the band matrix.
    for (int i = tid; i < PP_LEN; i += 256) {
        int j = i - 16;
        pp[i] = (j >= 0 && j < AR_P) ? params[j] : 0.0f;
    }
    __syncthreads();

    const int wave = tid >> 5;              // wave32 on gfx1250
    const int lane = tid & 31;
    const int half = lane >> 4;             // 0: lanes 0-15, 1: lanes 16-31
    const int lm   = lane & 15;             // M (for A) or N (for B/D)
    const int koff = half * 2;              // K sub-offset per ISA 16x4 layout

    // A band into registers once: areg[c].{x,y} = A[lm, 4c+koff+{0,1}]
    v2f areg[KCHUNKS];
#pragma unroll
    for (int c = 0; c < KCHUNKS; ++c) {
        int idx = 4 * c + koff - lm + 16;   // in [1, 159]
        areg[c].x = pp[idx];
        areg[c].y = pp[idx + 1];
    }

    const int off  = wave * TILE_OUT;       // this wave's tile inside LDS window
    const int bcol = 16 * lm;               // B column stride (N index)

    v8f acc = {};
#pragma unroll
    for (int c = 0; c < KCHUNKS; ++c) {
        const int k = 4 * c + koff;
        v2f b;
        b.x = ylds[off + bcol + k];
        b.y = ylds[off + bcol + k + 1];
        acc = __builtin_amdgcn_wmma_f32_16x16x4_f32(
            /*neg_a=*/false, areg[c], /*neg_b=*/false, b,
            /*c_mod=*/(short)0, acc, /*reuse_a=*/false, /*reuse_b=*/false);
    }

    // D layout: VGPR r, lane -> M = r + 8*half, N = lm.
    // This lane owns pred[A0 .. A0+7] contiguously (A0 multiple of 8).
    const int A0 = blockBase + off + 16 * lm + 8 * half;
    if (A0 < n) {                           // n % 8 == 0 -> whole-octet validity
        float4 lo = {acc[0], acc[1], acc[2], acc[3]};
        float4 hi = {acc[4], acc[5], acc[6], acc[7]};
        *reinterpret_cast<float4*>(pred + A0)     = lo;
        *reinterpret_cast<float4*>(pred + A0 + 4) = hi;
    }
}

// ---------------------------------------------------------------------------
// Kernel 2: deterministic per-block partial sums of (pred[i] - y[i+P])^2
// ---------------------------------------------------------------------------
__global__ __launch_bounds__(256) void AR_sqerr(const float* __restrict__ pred,
                                                const float* __restrict__ y,
                                                float* __restrict__ part, int n) {
    __shared__ float red[256];
    const int tid  = threadIdx.x;
    const int base = blockIdx.x * BLOCK_OUT;

    float s = 0.0f;
    for (int i = tid; i < BLOCK_OUT; i += 256) {
        int g = base + i;
        if (g < n) {
            float d = pred[g] - y[g + AR_P];
            s += d * d;
        }
    }
    red[tid] = s;
    __syncthreads();
    for (int w = 128; w > 0; w >>= 1) {
        if (tid < w) red[tid] += red[tid + w];
        __syncthreads();
    }
    if (tid == 0) part[blockIdx.x] = red[0];
}

// ---------------------------------------------------------------------------
// Kernel 3: loss = sum(part)/n + sum(|params|)   (single block, deterministic)
// ---------------------------------------------------------------------------
__global__ __launch_bounds__(256) void AR_finalize(const float* __restrict__ part,
                                                   int nparts,
                                                   const float* __restrict__ params,
                                                   float* __restrict__ out_loss, int n) {
    __shared__ float red[256];
    const int tid = threadIdx.x;

    float s = 0.0f;
    for (int i = tid; i < nparts; i += 256) s += part[i];
    red[tid] = s;
    __syncthreads();
    for (int w = 128; w > 0; w >>= 1) {
        if (tid < w) red[tid] += red[tid + w];
        __syncthreads();
    }
    float sse = red[0];
    __syncthreads();

    float l = 0.0f;
    for (int i = tid; i < AR_P; i += 256) l += fabsf(params[i]);
    red[tid] = l;
    __syncthreads();
    for (int w = 128; w > 0; w >>= 1) {
        if (tid < w) red[tid] += red[tid + w];
        __syncthreads();
    }
    if (tid == 0) out_loss[0] = sse / (float)n + red[0];
}

// ---------------------------------------------------------------------------
extern "C" void kernel_launch(void* const* d_in, const int* in_sizes, int n_in,
                              void* d_out, int out_size, void* d_ws, size_t ws_size,
                              hipStream_t stream) {
    const float* y      = (const float*)d_in[0];   // (N,) f32
    const float* params = (const float*)d_in[1];   // (P,) f32
    float*       pred   = (float*)d_out;           // pred (n) ++ loss (1)

    const int N = in_sizes[0];
    const int n = N - AR_P;                        // 2,000,000
    const int nblk = (n + BLOCK_OUT - 1) / BLOCK_OUT;

    float* part = (float*)d_ws;                    // nblk floats (~4 KB)

    AR_wmma_pred<<<nblk, 256, 0, stream>>>(y, params, pred, n, N);
    AR_sqerr   <<<nblk, 256, 0, stream>>>(pred, y, part, n);
    AR_finalize<<<1,    256, 0, stream>>>(part, nblk, params, pred + n, n);
}